// PillarFeatureNet_58737972740528
// MI455X (gfx1250) — compile-verified
//
#include <hip/hip_runtime.h>

// ---------------- problem constants ----------------
#define NXc   500
#define NYc   500
#define NCELL 250000
#define NPIL  12000
#define NPTS  150000
#define MAXP  100
#define SLOTS (NPIL*MAXP)
#define NB    2
#define NEG_BIG (-3.402823466e38f)

typedef __attribute__((ext_vector_type(2))) float v2f;
typedef __attribute__((ext_vector_type(8))) float v8f;

// ---------------- workspace layout (4-byte elements) ----------------
constexpr size_t OFF_CNT     = 0;                               // int [NB][NCELL]
constexpr size_t OFF_PMAP    = OFF_CNT     + (size_t)NB*NCELL;  // int [NB][NCELL]
constexpr size_t OFF_PILCELL = OFF_PMAP    + (size_t)NB*NCELL;  // int [NB][NPIL]
constexpr size_t OFF_PCNT    = OFF_PILCELL + (size_t)NB*NPIL;   // int [NB][NPIL]
constexpr size_t OFF_POFF    = OFF_PCNT    + (size_t)NB*NPIL;   // int [NB][NPIL+1]
constexpr size_t OFF_FILL    = OFF_POFF    + (size_t)NB*(NPIL+1);
constexpr size_t OFF_PTIDX   = OFF_FILL    + (size_t)NB*NPIL;   // int [NB][NPTS]
constexpr size_t OFF_SLOT    = OFF_PTIDX   + (size_t)NB*NPTS;   // int [NB][SLOTS]
constexpr size_t OFF_NUM     = OFF_SLOT    + (size_t)NB*SLOTS;  // int [NB]
constexpr size_t OFF_W1      = OFF_NUM     + (size_t)NB;        // float[64*8]
constexpr size_t OFF_B1      = OFF_W1      + 512;               // float[64]
constexpr size_t OFF_W2      = OFF_B1      + 64;                // float[64*64]
constexpr size_t OFF_B2      = OFF_W2      + 4096;              // float[64]

// ---------------- kernels ----------------

__global__ void fold_weights(const float* W1, const float* g1, const float* b1,
                             const float* rm1, const float* rv1,
                             const float* W2, const float* g2, const float* b2,
                             const float* rm2, const float* rv2,
                             float* weff1, float* bias1, float* weff2, float* bias2)
{
    int c = threadIdx.x;
    if (c >= 64) return;
    float s1 = g1[c] * rsqrtf(rv1[c] + 1e-5f);
    bias1[c] = b1[c] - rm1[c] * s1;
    for (int k = 0; k < 8; ++k)  weff1[c*8 + k] = W1[c*8 + k] * s1;
    float s2 = g2[c] * rsqrtf(rv2[c] + 1e-5f);
    bias2[c] = b2[c] - rm2[c] * s2;
    for (int k = 0; k < 64; ++k) weff2[c*64 + k] = W2[c*64 + k] * s2;
}

__global__ void init_ws(int* cnt, int* pcnt, int* fill, int* slotmap)
{
    int i = blockIdx.x * blockDim.x + threadIdx.x;
    if (i < NB*SLOTS) slotmap[i] = -1;
    if (i < NB*NCELL) cnt[i] = 0;
    if (i < NB*NPIL)  { pcnt[i] = 0; fill[i] = 0; }
}

__device__ __forceinline__ int cell_of(float x, float y, float z, bool& valid)
{
    valid = (x >= -40.0f) && (x < 40.0f) && (y >= -40.0f) && (y < 40.0f)
         && (z >= -3.0f) && (z < 1.0f);
    if (!valid) return 0;
    int xi = (int)floorf((x - (-40.0f)) / 0.16f);
    int yi = (int)floorf((y - (-40.0f)) / 0.16f);
    if ((unsigned)xi >= 500u || (unsigned)yi >= 500u) { valid = false; return 0; }
    return xi * NYc + yi;
}

__global__ void count_points(const float* pts, int* cnt)
{
    int gid = blockIdx.x * blockDim.x + threadIdx.x;
    if (gid >= NB*NPTS) return;
    const float* P = pts + (size_t)gid * 4;
    bool valid;
    int lin = cell_of(P[0], P[1], P[2], valid);
    if (!valid) return;
    int b = gid / NPTS;
    atomicAdd(&cnt[b*NCELL + lin], 1);
}

// single block of 1024 threads: rank cells (lin ascending), build pillar CSR offsets
__global__ void scan_cells(const int* cnt, int* pmap, int* pilcell, int* pcnt,
                           int* poff, int* numpil)
{
    __shared__ int sh[1024];
    __shared__ int carry_sh;
    int tid = threadIdx.x;
    for (int b = 0; b < NB; ++b) {
        if (tid == 0) carry_sh = 0;
        __syncthreads();
        const int* cb = cnt + b*NCELL;
        int* pm = pmap + b*NCELL;
        for (int base = 0; base < NCELL; base += 1024) {
            int cell = base + tid;
            int c = (cell < NCELL) ? cb[cell] : 0;
            int flag = (c > 0) ? 1 : 0;
            sh[tid] = flag; __syncthreads();
            for (int o = 1; o < 1024; o <<= 1) {
                int v = (tid >= o) ? sh[tid - o] : 0;
                __syncthreads();
                sh[tid] += v; __syncthreads();
            }
            int inc = sh[tid];
            int total = sh[1023];
            int carry = carry_sh;
            __syncthreads();
            int rank = carry + inc - flag;
            if (cell < NCELL) {
                int r = -1;
                if (flag && rank < NPIL) {
                    r = rank;
                    pilcell[b*NPIL + rank] = cell;
                    pcnt[b*NPIL + rank] = c;
                }
                pm[cell] = r;
            }
            if (tid == 0) carry_sh = carry + total;
            __syncthreads();
        }
        if (tid == 0) numpil[b] = (carry_sh < NPIL) ? carry_sh : NPIL;
        __syncthreads();

        // exclusive scan of pillar counts -> CSR offsets
        if (tid == 0) carry_sh = 0;
        __syncthreads();
        for (int base = 0; base < NPIL; base += 1024) {
            int p = base + tid;
            int v = (p < NPIL) ? pcnt[b*NPIL + p] : 0;
            sh[tid] = v; __syncthreads();
            for (int o = 1; o < 1024; o <<= 1) {
                int t = (tid >= o) ? sh[tid - o] : 0;
                __syncthreads();
                sh[tid] += t; __syncthreads();
            }
            int inc = sh[tid];
            int total = sh[1023];
            int carry = carry_sh;
            __syncthreads();
            if (p < NPIL) poff[b*(NPIL+1) + p] = carry + inc - v;
            if (tid == 0) carry_sh = carry + total;
            __syncthreads();
        }
        if (tid == 0) poff[b*(NPIL+1) + NPIL] = carry_sh;
        __syncthreads();
    }
}

__global__ void fill_csr(const float* pts, const int* pmap, const int* poff,
                         int* fill, int* ptidx)
{
    int gid = blockIdx.x * blockDim.x + threadIdx.x;
    if (gid >= NB*NPTS) return;
    const float* P = pts + (size_t)gid * 4;
    bool valid;
    int lin = cell_of(P[0], P[1], P[2], valid);
    if (!valid) return;
    int b = gid / NPTS, i = gid % NPTS;
    int p = pmap[b*NCELL + lin];
    if (p < 0) return;
    int slot = atomicAdd(&fill[b*NPIL + p], 1);
    ptidx[b*NPTS + poff[b*(NPIL+1) + p] + slot] = i;
}

// within each pillar, rank points by (z desc, original idx asc); keep first 100
__global__ void assign_slots(const float* pts, const int* poff, const int* ptidx,
                             const int* numpil, int* slotmap)
{
    int blk = blockIdx.x;
    int b = blk / NPIL, p = blk % NPIL;
    if (p >= numpil[b]) return;
    const int* off = poff + b*(NPIL+1);
    int s0 = off[p], n = off[p+1] - s0;
    const int* idxs = ptidx + b*NPTS + s0;
    const float* P = pts + (size_t)b*NPTS*4;
    for (int j = threadIdx.x; j < n; j += blockDim.x) {
        int ij = idxs[j];
        float zj = P[(size_t)ij*4 + 2];
        int pos = 0;
        for (int k = 0; k < n; ++k) {
            int ik = idxs[k];
            float zk = P[(size_t)ik*4 + 2];
            pos += ((zk > zj) || (zk == zj && ik < ij)) ? 1 : 0;
        }
        if (pos < MAXP) slotmap[(size_t)b*SLOTS + p*MAXP + pos] = ij;
    }
}

// one block per pillar: fused 8->64 and 64->64 MLP via V_WMMA_F32_16X16X4_F32 + maxpool
__global__ __launch_bounds__(128)
void pillar_mlp(const float* pts, const int* pilcell, const int* slotmap,
                const int* numpil, const float* weff1, const float* bias1,
                const float* weff2, const float* bias2,
                float* out_feat, float* out_coord)
{
    __shared__ float sW2[64*68];      // padded stride 68 -> conflict-free
    __shared__ float sW1[64*9];
    __shared__ float sB1[64];
    __shared__ float sB2[64];
    __shared__ float sH1[4][16*68];   // per-wave h1 tile, stride 68
    __shared__ float sMax[8][64];

    int blk = blockIdx.x;
    int b = blk / NPIL, p = blk % NPIL;
    int num = numpil[b];
    size_t fo = ((size_t)b*NPIL + p) * 64;
    size_t co = ((size_t)b*NPIL + p) * 3;
    int tid = threadIdx.x;

    if (p >= num) {   // uniform per block
        for (int t = tid; t < 64; t += 128) out_feat[fo + t] = 0.0f;
        if (tid == 0) { out_coord[co] = 0.0f; out_coord[co+1] = 0.0f; out_coord[co+2] = 0.0f; }
        return;
    }

    for (int i = tid; i < 64*64; i += 128) { int r = i >> 6, c = i & 63; sW2[r*68 + c] = weff2[i]; }
    for (int i = tid; i < 64*8;  i += 128) { int r = i >> 3, c = i & 7;  sW1[r*9  + c] = weff1[i]; }
    if (tid < 64) { sB1[tid] = bias1[tid]; sB2[tid] = bias2[tid]; }
    __syncthreads();

    int cell = pilcell[b*NPIL + p];
    int px = cell / NYc, py = cell % NYc;
    float xc = __fadd_rn(__fadd_rn(__fmul_rn((float)px, 0.16f), -40.0f), 0.08f);
    float yc = __fadd_rn(__fadd_rn(__fmul_rn((float)py, 0.16f), -40.0f), 0.08f);

    int wave = tid >> 5;
    int lane = tid & 31;
    int n0 = lane & 15;      // col for B/C/D frags; row for A frags
    int kh = lane >> 4;      // K-half for A/B frags; row-half for C/D frags
    const int* smap = slotmap + (size_t)b*SLOTS + p*MAXP;
    const float* P = pts + (size_t)b*NPTS*4;
    float* h1w = &sH1[wave][0];

    float pmx0 = NEG_BIG, pmx1 = NEG_BIG, pmx2 = NEG_BIG, pmx3 = NEG_BIG;

    for (int mt = wave; mt < 7; mt += 4) {           // 7 M-tiles cover 112 >= 100 slots
        // ---- build 8-dim features for row m = n0 of this tile ----
        int slot = mt*16 + n0;
        float f0=0,f1=0,f2=0,f3=0,f4=0,f5=0,f6=0,f7=0;
        if (slot < MAXP) {
            int idx = smap[slot];
            float x=0, y=0, z=0, it=0;
            if (idx >= 0) {
                const float4 q = *(const float4*)(P + (size_t)idx*4);
                x = q.x; y = q.y; z = q.z; it = q.w;
            }
            f0 = x; f1 = y; f2 = z; f3 = it;
            f4 = xc; f5 = yc; f6 = x - xc; f7 = y - yc;
        }

        // ---- layer 1: 8 -> 64, two K=4 WMMA steps per N-tile ----
        for (int nt = 0; nt < 4; ++nt) {
            int ch = nt*16 + n0;
            float bias = sB1[ch];
            v8f acc;
            for (int r = 0; r < 8; ++r) acc[r] = bias;
            v2f a0, b0, a1, b1v;
            a0.x = kh ? f2 : f0;  a0.y = kh ? f3 : f1;
            b0.x = sW1[ch*9 + 2*kh];     b0.y = sW1[ch*9 + 2*kh + 1];
            acc = __builtin_amdgcn_wmma_f32_16x16x4_f32(false, a0, false, b0, (short)0, acc, false, false);
            a1.x = kh ? f6 : f4;  a1.y = kh ? f7 : f5;
            b1v.x = sW1[ch*9 + 4 + 2*kh]; b1v.y = sW1[ch*9 + 5 + 2*kh];
            acc = __builtin_amdgcn_wmma_f32_16x16x4_f32(false, a1, false, b1v, (short)0, acc, false, false);
            for (int r = 0; r < 8; ++r) {
                float v = fmaxf(acc[r], 0.0f);           // BN folded, ReLU
                h1w[(r + 8*kh)*68 + ch] = v;             // row-major h1, padded stride
            }
        }

        // ---- layer 2: 64 -> 64, 16 K=4 WMMA steps per N-tile ----
        for (int nt = 0; nt < 4; ++nt) {
            int ch = nt*16 + n0;
            float bias = sB2[ch];
            v8f acc;
            for (int r = 0; r < 8; ++r) acc[r] = bias;
            for (int kk = 0; kk < 16; ++kk) {
                int kb = 4*kk + 2*kh;
                v2f a = *(const v2f*)&h1w[n0*68 + kb];    // A row = n0
                v2f bb = *(const v2f*)&sW2[ch*68 + kb];
                acc = __builtin_amdgcn_wmma_f32_16x16x4_f32(false, a, false, bb, (short)0, acc, false, false);
            }
            float mx = NEG_BIG;
            for (int r = 0; r < 8; ++r) {
                float v = fmaxf(acc[r], 0.0f);
                int rslot = mt*16 + r + 8*kh;
                v = (rslot < MAXP) ? v : NEG_BIG;        // exclude pad rows 100..111
                mx = fmaxf(mx, v);
            }
            if      (nt == 0) pmx0 = fmaxf(pmx0, mx);
            else if (nt == 1) pmx1 = fmaxf(pmx1, mx);
            else if (nt == 2) pmx2 = fmaxf(pmx2, mx);
            else              pmx3 = fmaxf(pmx3, mx);
        }
    }

    int row = wave*2 + kh;
    sMax[row][ 0 + n0] = pmx0;
    sMax[row][16 + n0] = pmx1;
    sMax[row][32 + n0] = pmx2;
    sMax[row][48 + n0] = pmx3;
    __syncthreads();

    if (tid < 64) {
        float m = sMax[0][tid];
        for (int r = 1; r < 8; ++r) m = fmaxf(m, sMax[r][tid]);
        out_feat[fo + tid] = m;
    }
    if (tid == 0) {
        out_coord[co]     = 0.0f;
        out_coord[co + 1] = (float)px;
        out_coord[co + 2] = (float)py;
    }
}

// ---------------- launch ----------------
extern "C" void kernel_launch(void* const* d_in, const int* in_sizes, int n_in,
                              void* d_out, int out_size, void* d_ws, size_t ws_size,
                              hipStream_t stream)
{
    const float* pts = (const float*)d_in[0];
    const float* W1  = (const float*)d_in[1];
    const float* g1  = (const float*)d_in[2];
    const float* b1  = (const float*)d_in[3];
    const float* rm1 = (const float*)d_in[4];
    const float* rv1 = (const float*)d_in[5];
    const float* W2  = (const float*)d_in[6];
    const float* g2  = (const float*)d_in[7];
    const float* b2  = (const float*)d_in[8];
    const float* rm2 = (const float*)d_in[9];
    const float* rv2 = (const float*)d_in[10];

    float* out_feat  = (float*)d_out;                     // [2][12000][64]
    float* out_coord = (float*)d_out + (size_t)NB*NPIL*64; // [2][12000][3]

    int*   W  = (int*)d_ws;
    int*   cnt     = W + OFF_CNT;
    int*   pmap    = W + OFF_PMAP;
    int*   pilcell = W + OFF_PILCELL;
    int*   pcnt    = W + OFF_PCNT;
    int*   poff    = W + OFF_POFF;
    int*   fill    = W + OFF_FILL;
    int*   ptidx   = W + OFF_PTIDX;
    int*   slotmap = W + OFF_SLOT;
    int*   numpil  = W + OFF_NUM;
    float* weff1   = (float*)(W + OFF_W1);
    float* bias1   = (float*)(W + OFF_B1);
    float* weff2   = (float*)(W + OFF_W2);
    float* bias2   = (float*)(W + OFF_B2);

    fold_weights<<<1, 64, 0, stream>>>(W1, g1, b1, rm1, rv1, W2, g2, b2, rm2, rv2,
                                       weff1, bias1, weff2, bias2);
    {
        int n = NB*SLOTS;
        init_ws<<<(n + 255)/256, 256, 0, stream>>>(cnt, pcnt, fill, slotmap);
    }
    {
        int n = NB*NPTS;
        count_points<<<(n + 255)/256, 256, 0, stream>>>(pts, cnt);
    }
    scan_cells<<<1, 1024, 0, stream>>>(cnt, pmap, pilcell, pcnt, poff, numpil);
    {
        int n = NB*NPTS;
        fill_csr<<<(n + 255)/256, 256, 0, stream>>>(pts, pmap, poff, fill, ptidx);
    }
    assign_slots<<<NB*NPIL, 64, 0, stream>>>(pts, poff, ptidx, numpil, slotmap);
    pillar_mlp<<<NB*NPIL, 128, 0, stream>>>(pts, pilcell, slotmap, numpil,
                                            weff1, bias1, weff2, bias2,
                                            out_feat, out_coord);
}